// MultiHeadDotProduct_69853348102549
// MI455X (gfx1250) — compile-verified
//
#include <hip/hip_runtime.h>

typedef __attribute__((ext_vector_type(2))) float v2f;
typedef __attribute__((ext_vector_type(8))) float v8f;

#define DD   128   // feature dim
#define HH   8     // heads
#define HDD  16    // head dim
#define LDSW 132   // padded LDS row stride (floats): 132*4=528B -> 16B aligned, bank-spread

// ---------------------------------------------------------------------------
// out[n,:] = X[n,:] @ W^T + b       (X: nrows x 128, W: 128x128 row-major)
// 4 waves / block, each wave computes a 16x128 strip with f32 WMMA.
// ---------------------------------------------------------------------------
__global__ __launch_bounds__(128) void gemm_xwt_bias(
    const float* __restrict__ X, const float* __restrict__ W,
    const float* __restrict__ bias, float* __restrict__ out, int nrows)
{
  __shared__ float sW[DD * LDSW];        // 67.5 KB
  __shared__ float sA[4 * 16 * LDSW];    // 33.8 KB (one 16x128 tile per wave)
  __shared__ float sBias[DD];

  const int tid = threadIdx.x;
  // Stage W into LDS (coalesced float4, padded stride).
  for (int i = tid; i < DD * (DD / 4); i += 128) {
    const int r  = i >> 5;
    const int c4 = (i & 31) << 2;
    const float4 w = *(const float4*)(W + r * DD + c4);
    *(float4*)&sW[r * LDSW + c4] = w;
  }
  sBias[tid] = bias[tid];
  __syncthreads();

  const int wave = tid >> 5;
  const int lane = tid & 31;
  const int row0 = (blockIdx.x * 4 + wave) * 16;
  if (row0 < nrows) {                       // wave-uniform: EXEC all-ones inside
    float* sAw = &sA[wave * 16 * LDSW];
    // Stage this wave's 16x128 A tile (coalesced float4; same-wave DS ordering).
    #pragma unroll
    for (int i = 0; i < 16; ++i) {
      const int chunk = i * 32 + lane;      // 512 float4 chunks
      const int r  = chunk >> 5;
      const int c4 = (chunk & 31) << 2;
      *(float4*)&sAw[r * LDSW + c4] =
          *(const float4*)(X + (size_t)(row0 + r) * DD + c4);
    }
    const int m    = lane & 15;             // row (A) / col (B,D) within tile
    const int half = lane >> 4;             // K-pair select (A,B); row-half (D)
    v8f acc[8] = {};
    #pragma unroll 4
    for (int kk = 0; kk < DD; kk += 4) {
      // A 16x4 f32 frag: VGPR0=K(0|2), VGPR1=K(1|3) per ISA layout
      v2f a;
      a.x = sAw[m * LDSW + kk + 2 * half];
      a.y = sAw[m * LDSW + kk + 2 * half + 1];
      #pragma unroll
      for (int nt = 0; nt < 8; ++nt) {
        // B[k][n] = W[n][k]; lanes span n, VGPRs span k
        v2f b;
        b.x = sW[(nt * 16 + m) * LDSW + kk + 2 * half];
        b.y = sW[(nt * 16 + m) * LDSW + kk + 2 * half + 1];
        acc[nt] = __builtin_amdgcn_wmma_f32_16x16x4_f32(
            false, a, false, b, (short)0, acc[nt], false, false);
      }
    }
    // D layout: VGPR i -> row half*8+i, lane&15 -> col
    #pragma unroll
    for (int nt = 0; nt < 8; ++nt) {
      const float bv = sBias[nt * 16 + m];
      #pragma unroll
      for (int i = 0; i < 8; ++i)
        out[(size_t)(row0 + half * 8 + i) * DD + nt * 16 + m] = acc[nt][i] + bv;
    }
  }
}

// ---------------------------------------------------------------------------
// Edge phase: one thread per (edge, head).
// ---------------------------------------------------------------------------
__global__ __launch_bounds__(256) void edge_scores_kernel(
    const float* __restrict__ q, const float* __restrict__ k,
    const int* __restrict__ ei, float* __restrict__ scores,
    unsigned* __restrict__ smax, int EH)
{
  const int t = blockIdx.x * 256 + threadIdx.x;
  if (t >= EH) return;
  const int e = t >> 3, h = t & 7;
  const int r = ei[2 * e + 0];
  const int c = ei[2 * e + 1];
  const float4* qp = (const float4*)(q + (size_t)c * DD + h * HDD);
  const float4* kp = (const float4*)(k + (size_t)r * DD + h * HDD);
  float s = 0.f;
  #pragma unroll
  for (int i = 0; i < 4; ++i) {
    const float4 a = qp[i], b = kp[i];
    s += a.x * b.x + a.y * b.y + a.z * b.z + a.w * b.w;
  }
  s *= 0.25f;                               // 1/sqrt(HD)
  scores[t] = s;
  // smax is pre-zeroed => already clamped at 0; positive f32 order == uint order
  if (s > 0.f) atomicMax(&smax[c * HH + h], __float_as_uint(s));
}

__global__ __launch_bounds__(256) void edge_exp_kernel(
    const int* __restrict__ ei, const unsigned* __restrict__ smax,
    float* __restrict__ scores, float* __restrict__ denom,
    float* __restrict__ cnt, int EH)
{
  const int t = blockIdx.x * 256 + threadIdx.x;
  if (t >= EH) return;
  const int e = t >> 3, h = t & 7;
  const int c = ei[2 * e + 1];
  const float sm = __uint_as_float(smax[c * HH + h]);
  const float ex = __expf(scores[t] - sm);
  scores[t] = ex;                           // overwrite score with exp in place
  atomicAdd(&denom[c * HH + h], ex);
  if (h == 0) atomicAdd(&cnt[c], 1.0f);
}

__global__ __launch_bounds__(256) void edge_agg_kernel(
    const int* __restrict__ ei, const float* __restrict__ v,
    const float* __restrict__ scores, const unsigned* __restrict__ smax,
    const float* __restrict__ denom, float* __restrict__ agg, int EH)
{
  const int t = blockIdx.x * 256 + threadIdx.x;
  if (t >= EH) return;
  const int e = t >> 3, h = t & 7;
  const int r = ei[2 * e + 0];
  const int c = ei[2 * e + 1];
  const float sm   = __uint_as_float(smax[c * HH + h]);
  const float attn = scores[t] / (denom[c * HH + h] + __expf(-sm));
  const float* vp = v + (size_t)r * DD + h * HDD;
  float* ap = agg + (size_t)c * DD + h * HDD;
  #pragma unroll
  for (int d = 0; d < HDD; ++d) atomicAdd(&ap[d], attn * vp[d]);
}

__global__ __launch_bounds__(256) void norm_kernel(
    float* __restrict__ agg, const float* __restrict__ cnt, int ND)
{
  const int t = blockIdx.x * 256 + threadIdx.x;
  if (t >= ND) return;
  const int n = t >> 7;                     // /128
  agg[t] *= 1.0f / fmaxf(cnt[n], 1.0f);
}

// ---------------------------------------------------------------------------
extern "C" void kernel_launch(void* const* d_in, const int* in_sizes, int n_in,
                              void* d_out, int out_size, void* d_ws, size_t ws_size,
                              hipStream_t stream)
{
  const float* feats = (const float*)d_in[0];
  const int*   ei    = (const int*)  d_in[1];   // edge_index (harness int convention)
  const float* Wq = (const float*)d_in[2]; const float* bq = (const float*)d_in[3];
  const float* Wk = (const float*)d_in[4]; const float* bk = (const float*)d_in[5];
  const float* Wv = (const float*)d_in[6]; const float* bv = (const float*)d_in[7];
  const float* Wo = (const float*)d_in[8]; const float* bo = (const float*)d_in[9];
  float* outp = (float*)d_out;

  const int N = in_sizes[0] / DD;
  const int E = in_sizes[1] / 2;
  const int EH = E * HH;

  // Workspace layout (floats): q | k | v | scores | agg | smax | denom | cnt
  float* q      = (float*)d_ws;
  float* k      = q      + (size_t)N * DD;
  float* v      = k      + (size_t)N * DD;
  float* scores = v      + (size_t)N * DD;
  float* agg    = scores + (size_t)E * HH;
  float* smax   = agg    + (size_t)N * DD;
  float* denom  = smax   + (size_t)N * HH;
  float* cnt    = denom  + (size_t)N * HH;

  // Zero agg..cnt in one shot (smax=0.0f == clamp-at-zero init).
  const size_t zero_bytes =
      ((size_t)N * DD + 2 * (size_t)N * HH + (size_t)N) * sizeof(float);
  hipMemsetAsync(agg, 0, zero_bytes, stream);

  const int tiles = (N + 15) / 16;
  dim3 ggrid((tiles + 3) / 4);
  gemm_xwt_bias<<<ggrid, 128, 0, stream>>>(feats, Wq, bq, q, N);
  gemm_xwt_bias<<<ggrid, 128, 0, stream>>>(feats, Wk, bk, k, N);
  gemm_xwt_bias<<<ggrid, 128, 0, stream>>>(feats, Wv, bv, v, N);

  const int eblocks = (EH + 255) / 256;
  edge_scores_kernel<<<eblocks, 256, 0, stream>>>(q, k, ei, scores,
                                                  (unsigned*)smax, EH);
  edge_exp_kernel<<<eblocks, 256, 0, stream>>>(ei, (const unsigned*)smax,
                                               scores, denom, cnt, EH);
  edge_agg_kernel<<<eblocks, 256, 0, stream>>>(ei, v, scores,
                                               (const unsigned*)smax, denom,
                                               agg, EH);
  const int ND = N * DD;
  norm_kernel<<<(ND + 255) / 256, 256, 0, stream>>>(agg, cnt, ND);

  gemm_xwt_bias<<<ggrid, 128, 0, stream>>>(agg, Wo, bo, outp, N);
}